// VSSBlock_47510928228777
// MI455X (gfx1250) — compile-verified
//
#include <hip/hip_runtime.h>
#include <hip/hip_bf16.h>
#include <math.h>

#define D_MODEL 96
#define D_STATE 128
#define D_CONVK 4
#define D_INNER 192
#define NHEADS 3
#define HEADDIM 64
#define CHUNK 256
#define CONV_DIM 448
#define D_IN_PROJ 643
#define SEQLEN 4096
#define NSEQ 8
#define NCHUNK 16
#define MTOT (NSEQ*SEQLEN)   /* 32768 rows */

typedef __attribute__((ext_vector_type(2))) float v2f;
typedef __attribute__((ext_vector_type(8))) float v8f;

static __device__ __forceinline__ v8f wmma4(v2f a, v2f b, v8f c) {
  // D = A(16x4,f32) x B(4x16,f32) + C(16x16,f32) -> v_wmma_f32_16x16x4_f32
  return __builtin_amdgcn_wmma_f32_16x16x4_f32(false, a, false, b, (short)0, c,
                                               false, false);
}
static __device__ __forceinline__ v2f ld2(const float* p) {
  return *(const v2f*)p;      // 8B-aligned by construction -> global_load_b64 / ds_load_b64
}
static __device__ __forceinline__ float siluf(float x) {
  return x / (1.0f + __expf(-x));
}

// ---------------------------------------------------------------- K1: RMSNorm + cross-scan
__global__ __launch_bounds__(128) void k1_rms_scan(const float* __restrict__ x,
                                                   const float* __restrict__ norm_w,
                                                   float* __restrict__ xs) {
  int bid = blockIdx.x;          // (b, pixel): 0..8191
  int b = bid >> 12;
  int l = bid & 4095;
  int h = l >> 6, w = l & 63;
  int t = threadIdx.x;
  __shared__ float red[128];
  float v = 0.f;
  if (t < D_MODEL) v = x[((b * D_MODEL + t) * 64 + h) * 64 + w];
  red[t] = v * v;
  __syncthreads();
  for (int s = 64; s > 0; s >>= 1) { if (t < s) red[t] += red[t + s]; __syncthreads(); }
  float scale = rsqrtf(red[0] / (float)D_MODEL + 1.1920929e-07f);
  if (t < D_MODEL) {
    float xn = v * scale * norm_w[t];
    int l0 = h * 64 + w;
    int l1 = (63 - h) * 64 + (63 - w);
    int l2 = w * 64 + h;
    int l3 = (63 - w) * 64 + (63 - h);
    xs[((4 * b + 0) * SEQLEN + l0) * D_MODEL + t] = xn;
    xs[((4 * b + 1) * SEQLEN + l1) * D_MODEL + t] = xn;
    xs[((4 * b + 2) * SEQLEN + l2) * D_MODEL + t] = xn;
    xs[((4 * b + 3) * SEQLEN + l3) * D_MODEL + t] = xn;
  }
}

// ---------------------------------------------------------------- K2: in_proj GEMM (WMMA f32)
#define NTILE_INPROJ 41          /* ceil(643/16) */
__global__ __launch_bounds__(256) void k2_inproj(const float* __restrict__ xs,
                                                 const float* __restrict__ Wp,
                                                 float* __restrict__ Z,
                                                 float* __restrict__ XBCraw,
                                                 float* __restrict__ DTraw) {
  int wave = blockIdx.x * (blockDim.x >> 5) + (threadIdx.x >> 5);
  int lane = threadIdx.x & 31;
  int lo = lane & 15, hi = lane >> 4;
  int mt = wave / NTILE_INPROJ;
  int nt = wave % NTILE_INPROJ;
  if (mt >= MTOT / 16) return;
  int mBase = mt * 16;
  int n = nt * 16 + lo;
  int nc = n < D_IN_PROJ ? n : (D_IN_PROJ - 1);   // clamp: padded cols never stored
  const float* arow = xs + (mBase + lo) * D_MODEL;
  const float* brow = Wp + nc * D_MODEL;
  v8f acc = {};
#pragma unroll
  for (int k0 = 0; k0 < D_MODEL; k0 += 4) {
    int ka = k0 + 2 * hi;
    v2f a = ld2(arow + ka);
    v2f b = ld2(brow + ka);
    acc = wmma4(a, b, acc);
  }
  if (n >= D_IN_PROJ) return;
#pragma unroll
  for (int r = 0; r < 8; ++r) {
    int m = mBase + r + 8 * hi;
    float vv = acc[r];
    if (n < D_INNER)                 Z[m * D_INNER + n] = vv;
    else if (n < D_INNER + CONV_DIM) XBCraw[m * CONV_DIM + (n - D_INNER)] = vv;
    else                             DTraw[m * NHEADS + (n - D_INNER - CONV_DIM)] = vv;
  }
}

// ---------------------------------------------------------------- K3: depthwise conv + SiLU
__global__ __launch_bounds__(256) void k3_conv(const float* __restrict__ XBCraw,
                                               const float* __restrict__ conv_w,
                                               const float* __restrict__ conv_b,
                                               float* __restrict__ XBCc) {
  int idx = blockIdx.x * 256 + threadIdx.x;
  if (idx >= MTOT * CONV_DIM) return;
  int c = idx % CONV_DIM;
  int m = idx / CONV_DIM;
  int l = m & (SEQLEN - 1);
  float s = conv_b[c];
#pragma unroll
  for (int i = 0; i < D_CONVK; ++i) {
    int ls = l - 3 + i;
    if (ls >= 0) s += conv_w[c * D_CONVK + i] * XBCraw[(m - 3 + i) * CONV_DIM + c];
  }
  XBCc[m * CONV_DIM + c] = siluf(s);
}

// ---------------------------------------------------------------- K3b: dt = softplus(raw + bias)
__global__ __launch_bounds__(256) void k3b_dt(const float* __restrict__ DTraw,
                                              const float* __restrict__ dt_bias,
                                              float* __restrict__ DT) {
  int idx = blockIdx.x * 256 + threadIdx.x;
  if (idx >= MTOT * NHEADS) return;
  int hh = idx % NHEADS;
  float v = DTraw[idx] + dt_bias[hh];
  DT[idx] = (v > 20.f) ? v : log1pf(__expf(v));
}

// ---------------------------------------------------------------- K4: per-chunk cumsum + wt
__global__ __launch_bounds__(128) void k4_cumsum(const float* __restrict__ DT,
                                                 const float* __restrict__ A_log,
                                                 float* __restrict__ dAc,
                                                 float* __restrict__ WT) {
  int idx = blockIdx.x * 128 + threadIdx.x;
  if (idx >= NSEQ * NCHUNK * NHEADS) return;
  int hh = idx % NHEADS;
  int ck = (idx / NHEADS) % NCHUNK;
  int seq = idx / (NHEADS * NCHUNK);
  float A = -__expf(A_log[hh]);
  int base = seq * SEQLEN + ck * CHUNK;
  float acc = 0.f;
  for (int s = 0; s < CHUNK; ++s) {
    acc += DT[(base + s) * NHEADS + hh] * A;
    dAc[(base + s) * NHEADS + hh] = acc;
  }
  float last = acc;
  for (int s = 0; s < CHUNK; ++s) {
    int m = (base + s) * NHEADS + hh;
    WT[m] = __expf(last - dAc[m]) * DT[m];
  }
}

// ---------------------------------------------------------------- K5: chunk states (WMMA f32)
__global__ __launch_bounds__(512) void k5_states(const float* __restrict__ XBCc,
                                                 const float* __restrict__ WT,
                                                 float* __restrict__ states) {
  int blk = blockIdx.x;                           // (seq,chunk,head)
  int hh = blk % NHEADS;
  int ck = (blk / NHEADS) % NCHUNK;
  int seq = blk / (NHEADS * NCHUNK);
  int base = seq * SEQLEN + ck * CHUNK;
  int wave = threadIdx.x >> 5;
  int lane = threadIdx.x & 31;
  int lo = lane & 15, hi = lane >> 4;
  float* out = states + (((seq * NCHUNK + ck) * NHEADS) + hh) * HEADDIM * D_STATE;
#pragma unroll
  for (int rep = 0; rep < 2; ++rep) {
    int tile = wave + 16 * rep;                   // 0..31
    int pTile = tile >> 3;                        // 0..3 (p)
    int nTile = tile & 7;                         // 0..7 (n)
    int p = pTile * 16 + lo;
    int n = nTile * 16 + lo;
    v8f acc = {};
#pragma unroll 4
    for (int k0 = 0; k0 < CHUNK; k0 += 4) {
      int t0 = base + k0 + 2 * hi;
      v2f a, b;
      a.x = XBCc[t0 * CONV_DIM + hh * HEADDIM + p] * WT[t0 * NHEADS + hh];
      a.y = XBCc[(t0 + 1) * CONV_DIM + hh * HEADDIM + p] * WT[(t0 + 1) * NHEADS + hh];
      b.x = XBCc[t0 * CONV_DIM + D_INNER + n];
      b.y = XBCc[(t0 + 1) * CONV_DIM + D_INNER + n];
      acc = wmma4(a, b, acc);
    }
#pragma unroll
    for (int r = 0; r < 8; ++r)
      out[(pTile * 16 + r + 8 * hi) * D_STATE + nTile * 16 + lo] = acc[r];
  }
}

// ---------------------------------------------------------------- K6: inter-chunk scan
__global__ __launch_bounds__(256) void k6_scan(const float* __restrict__ states,
                                               const float* __restrict__ dAc,
                                               float* __restrict__ prev) {
  int blk = blockIdx.x;                           // seq*3 + head, 0..23
  int hh = blk % NHEADS;
  int seq = blk / NHEADS;
  int tid = threadIdx.x;
  float hreg[32];
#pragma unroll
  for (int j = 0; j < 32; ++j) hreg[j] = 0.f;
  for (int ck = 0; ck < NCHUNK; ++ck) {
    float decay = __expf(dAc[(seq * SEQLEN + ck * CHUNK + CHUNK - 1) * NHEADS + hh]);
    const float* st = states + ((seq * NCHUNK + ck) * NHEADS + hh) * HEADDIM * D_STATE;
    float* pv = prev + ((seq * NCHUNK + ck) * NHEADS + hh) * HEADDIM * D_STATE;
#pragma unroll
    for (int j = 0; j < 32; ++j) {
      int e = j * 256 + tid;
      pv[e] = hreg[j];
      hreg[j] = decay * hreg[j] + st[e];
    }
  }
}

// ---------------------------------------------------------------- K7: fused SSD output (WMMA f32)
__global__ __launch_bounds__(512) void k7_y(const float* __restrict__ XBCc,
                                            const float* __restrict__ dAc,
                                            const float* __restrict__ DT,
                                            const float* __restrict__ prev,
                                            const float* __restrict__ Dparam,
                                            float* __restrict__ Y) {
  __shared__ float latt[16 * 256];                // 16 waves x 16x16 att tile
  int blk = blockIdx.x;                           // (seq,chunk,head)
  int hh = blk % NHEADS;
  int ck = (blk / NHEADS) % NCHUNK;
  int seq = blk / (NHEADS * NCHUNK);
  int base = seq * SEQLEN + ck * CHUNK;
  int sT = threadIdx.x >> 5;                      // wave id == s-tile
  int lane = threadIdx.x & 31;
  int lo = lane & 15, hi = lane >> 4;
  float* myatt = latt + sT * 256;
  const float* pv = prev + ((seq * NCHUNK + ck) * NHEADS + hh) * HEADDIM * D_STATE;

  float dAs_r[8];
#pragma unroll
  for (int r = 0; r < 8; ++r)
    dAs_r[r] = dAc[(base + sT * 16 + r + 8 * hi) * NHEADS + hh];

  v8f acc0 = {}, acc1 = {}, acc2 = {}, acc3 = {};

  // ---- inter-chunk: y2 = C @ prev^T  (M=s, K=n, N=p), then row-scale exp(dAc_s)
  {
    const float* Crow = XBCc + (base + sT * 16 + lo) * CONV_DIM + D_INNER + D_STATE;
#pragma unroll 4
    for (int k0 = 0; k0 < D_STATE; k0 += 4) {
      int ka = k0 + 2 * hi;
      v2f a = ld2(Crow + ka);
      v2f b0 = ld2(pv + (0 * 16 + lo) * D_STATE + ka);
      v2f b1 = ld2(pv + (1 * 16 + lo) * D_STATE + ka);
      v2f b2 = ld2(pv + (2 * 16 + lo) * D_STATE + ka);
      v2f b3 = ld2(pv + (3 * 16 + lo) * D_STATE + ka);
      acc0 = wmma4(a, b0, acc0);
      acc1 = wmma4(a, b1, acc1);
      acc2 = wmma4(a, b2, acc2);
      acc3 = wmma4(a, b3, acc3);
    }
  }
  {
    float er[8];
#pragma unroll
    for (int r = 0; r < 8; ++r) er[r] = __expf(dAs_r[r]);
#pragma unroll
    for (int r = 0; r < 8; ++r) {
      acc0[r] *= er[r]; acc1[r] *= er[r]; acc2[r] *= er[r]; acc3[r] *= er[r];
    }
  }

  // ---- intra-chunk causal attention
  for (int tT = 0; tT <= sT; ++tT) {
    // G = C_s @ B_t^T (M=s, K=n, N=t)
    v8f g = {};
    const float* Crow = XBCc + (base + sT * 16 + lo) * CONV_DIM + D_INNER + D_STATE;
    const float* Brow = XBCc + (base + tT * 16 + lo) * CONV_DIM + D_INNER;
#pragma unroll 4
    for (int k0 = 0; k0 < D_STATE; k0 += 4) {
      int ka = k0 + 2 * hi;
      v2f a = ld2(Crow + ka);
      v2f b = ld2(Brow + ka);
      g = wmma4(a, b, g);
    }
    // elementwise: mask * exp(dAc_s - dAc_t) * dt_t ; stage to LDS
    int tg = base + tT * 16 + lo;
    float dAt = dAc[tg * NHEADS + hh];
    float dtt = DT[tg * NHEADS + hh];
#pragma unroll
    for (int r = 0; r < 8; ++r) {
      int srow = sT * 16 + r + 8 * hi;
      int tcol = tT * 16 + lo;
      float val = (srow >= tcol) ? g[r] * __expf(dAs_r[r] - dAt) * dtt : 0.f;
      myatt[(r + 8 * hi) * 16 + lo] = val;
    }
    asm volatile("s_wait_dscnt 0" ::: "memory");
    // y += att @ xh_t  (M=s, K=t(16), N=p)
#pragma unroll
    for (int k0 = 0; k0 < 16; k0 += 4) {
      int ka = k0 + 2 * hi;
      v2f a = ld2(myatt + lo * 16 + ka);
      int trow = base + tT * 16 + ka;
      const float* xr0 = XBCc + trow * CONV_DIM + hh * HEADDIM;
      const float* xr1 = XBCc + (trow + 1) * CONV_DIM + hh * HEADDIM;
      v2f b0, b1, b2, b3;
      b0.x = xr0[0 * 16 + lo]; b0.y = xr1[0 * 16 + lo];
      b1.x = xr0[1 * 16 + lo]; b1.y = xr1[1 * 16 + lo];
      b2.x = xr0[2 * 16 + lo]; b2.y = xr1[2 * 16 + lo];
      b3.x = xr0[3 * 16 + lo]; b3.y = xr1[3 * 16 + lo];
      acc0 = wmma4(a, b0, acc0);
      acc1 = wmma4(a, b1, acc1);
      acc2 = wmma4(a, b2, acc2);
      acc3 = wmma4(a, b3, acc3);
    }
  }

  // ---- + D * xh, store
  float Dh = Dparam[hh];
#pragma unroll
  for (int r = 0; r < 8; ++r) {
    int sg = base + sT * 16 + r + 8 * hi;
    const float* xr = XBCc + sg * CONV_DIM + hh * HEADDIM;
    float* yo = Y + sg * D_INNER + hh * HEADDIM;
    yo[0 * 16 + lo] = acc0[r] + Dh * xr[0 * 16 + lo];
    yo[1 * 16 + lo] = acc1[r] + Dh * xr[1 * 16 + lo];
    yo[2 * 16 + lo] = acc2[r] + Dh * xr[2 * 16 + lo];
    yo[3 * 16 + lo] = acc3[r] + Dh * xr[3 * 16 + lo];
  }
}

// ---------------------------------------------------------------- K8a: gate + RMSNorm
__global__ __launch_bounds__(256) void k8a_gatenorm(const float* __restrict__ Y,
                                                    const float* __restrict__ Z,
                                                    const float* __restrict__ gnorm_w,
                                                    float* __restrict__ G) {
  int m = blockIdx.x;
  int t = threadIdx.x;
  __shared__ float red[256];
  float v = 0.f;
  if (t < D_INNER) {
    float z = Z[m * D_INNER + t];
    v = Y[m * D_INNER + t] * siluf(z);
  }
  red[t] = v * v;
  __syncthreads();
  for (int s = 128; s > 0; s >>= 1) { if (t < s) red[t] += red[t + s]; __syncthreads(); }
  float sc = rsqrtf(red[0] / (float)D_INNER + 1e-5f);
  if (t < D_INNER) G[m * D_INNER + t] = v * sc * gnorm_w[t];
}

// ---------------------------------------------------------------- K8b: out_proj GEMM (WMMA f32)
#define NTILE_OUT 6
__global__ __launch_bounds__(256) void k8b_outproj(const float* __restrict__ G,
                                                   const float* __restrict__ Wo,
                                                   float* __restrict__ ys) {
  int wave = blockIdx.x * (blockDim.x >> 5) + (threadIdx.x >> 5);
  int lane = threadIdx.x & 31;
  int lo = lane & 15, hi = lane >> 4;
  int mt = wave / NTILE_OUT, nt = wave % NTILE_OUT;
  if (mt >= MTOT / 16) return;
  int mBase = mt * 16;
  int n = nt * 16 + lo;                 // < 96 always
  const float* arow = G + (mBase + lo) * D_INNER;
  const float* brow = Wo + n * D_INNER;
  v8f acc = {};
#pragma unroll
  for (int k0 = 0; k0 < D_INNER; k0 += 4) {
    int ka = k0 + 2 * hi;
    v2f a = ld2(arow + ka);
    v2f b = ld2(brow + ka);
    acc = wmma4(a, b, acc);
  }
#pragma unroll
  for (int r = 0; r < 8; ++r) ys[(mBase + r + 8 * hi) * D_MODEL + n] = acc[r];
}

// ---------------------------------------------------------------- K8c: un-scan + residual
__global__ __launch_bounds__(256) void k8c_unscan(const float* __restrict__ ys,
                                                  const float* __restrict__ x,
                                                  float* __restrict__ out) {
  int idx = blockIdx.x * 256 + threadIdx.x;       // over 2*96*64*64
  if (idx >= 2 * D_MODEL * SEQLEN) return;
  int w = idx & 63;
  int hpix = (idx >> 6) & 63;
  int c = (idx >> 12) % D_MODEL;
  int b = idx / (D_MODEL * SEQLEN);
  int l0 = hpix * 64 + w;
  int l1 = (63 - hpix) * 64 + (63 - w);
  int l2 = w * 64 + hpix;
  int l3 = (63 - w) * 64 + (63 - hpix);
  float s = ys[((4 * b + 0) * SEQLEN + l0) * D_MODEL + c]
          + ys[((4 * b + 1) * SEQLEN + l1) * D_MODEL + c]
          + ys[((4 * b + 2) * SEQLEN + l2) * D_MODEL + c]
          + ys[((4 * b + 3) * SEQLEN + l3) * D_MODEL + c];
  out[idx] = 0.25f * s + x[idx];
}

// ---------------------------------------------------------------- launcher
extern "C" void kernel_launch(void* const* d_in, const int* in_sizes, int n_in,
                              void* d_out, int out_size, void* d_ws, size_t ws_size,
                              hipStream_t stream) {
  const float* x          = (const float*)d_in[0];
  const float* norm_w     = (const float*)d_in[1];
  const float* in_proj_w  = (const float*)d_in[2];
  const float* conv_w     = (const float*)d_in[3];
  const float* conv_b     = (const float*)d_in[4];
  const float* dt_bias    = (const float*)d_in[5];
  const float* A_log      = (const float*)d_in[6];
  const float* Dp         = (const float*)d_in[7];
  const float* gnorm_w    = (const float*)d_in[8];
  const float* out_proj_w = (const float*)d_in[9];
  float* out = (float*)d_out;

  float* ws     = (float*)d_ws;
  float* xs     = ws;                       // 3,145,728
  float* Z      = xs + 3145728;             // 6,291,456
  float* XBCraw = Z + 6291456;              // 14,680,064
  float* XBCc   = XBCraw + 14680064;        // 14,680,064
  float* DTraw  = XBCc + 14680064;          //    98,304
  float* DT     = DTraw + 98304;            //    98,304
  float* dAc    = DT + 98304;               //    98,304
  float* WT     = dAc + 98304;              //    98,304
  float* states = WT + 98304;               // 3,145,728
  float* prev   = states + 3145728;         // 3,145,728
  // aliases (source regions dead by the time these are written)
  float* Y   = XBCraw;                      // XBCraw dead after conv
  float* G   = XBCraw + 6291456;            // fits: 2*6,291,456 <= 14,680,064
  float* ysb = xs;                          // xs dead after in_proj

  k1_rms_scan<<<8192, 128, 0, stream>>>(x, norm_w, xs);
  k2_inproj<<<(2048 * NTILE_INPROJ) / 8, 256, 0, stream>>>(xs, in_proj_w, Z, XBCraw, DTraw);
  k3_conv<<<(MTOT * CONV_DIM) / 256, 256, 0, stream>>>(XBCraw, conv_w, conv_b, XBCc);
  k3b_dt<<<(MTOT * NHEADS) / 256, 256, 0, stream>>>(DTraw, dt_bias, DT);
  k4_cumsum<<<3, 128, 0, stream>>>(DT, A_log, dAc, WT);
  k5_states<<<NSEQ * NCHUNK * NHEADS, 512, 0, stream>>>(XBCc, WT, states);
  k6_scan<<<NSEQ * NHEADS, 256, 0, stream>>>(states, dAc, prev);
  k7_y<<<NSEQ * NCHUNK * NHEADS, 512, 0, stream>>>(XBCc, dAc, DT, prev, Dp, Y);
  k8a_gatenorm<<<MTOT, 256, 0, stream>>>(Y, Z, gnorm_w, G);
  k8b_outproj<<<(2048 * NTILE_OUT) / 8, 256, 0, stream>>>(G, out_proj_w, ysb);
  k8c_unscan<<<(2 * D_MODEL * SEQLEN) / 256, 256, 0, stream>>>(ysb, x, out);
}